// UnifiedGNN_56521769616171
// MI455X (gfx1250) — compile-verified
//
#include <hip/hip_runtime.h>
#include <hip/hip_bf16.h>

#define N_NODES 50000
#define DIM     128
#define N_EDGES 600000
#define LN_EPS  1e-5f

typedef __attribute__((ext_vector_type(2))) float v2f;
typedef __attribute__((ext_vector_type(8))) float v8f;

// ---------------------------------------------------------------------------
// Degree pipeline: deg starts at 1 (self loop), += 1 per in-edge, dis = rsqrt
// ---------------------------------------------------------------------------
__global__ void init_deg_kernel(float* __restrict__ deg) {
    int i = blockIdx.x * 256 + threadIdx.x;
    if (i < N_NODES) deg[i] = 1.0f;
}

__global__ void accum_deg_kernel(const int* __restrict__ dst, float* __restrict__ deg) {
    int e = blockIdx.x * 256 + threadIdx.x;
    if (e < N_EDGES) atomicAdd(&deg[dst[e]], 1.0f);
}

__global__ void finalize_deg_kernel(float* __restrict__ deg) {
    int i = blockIdx.x * 256 + threadIdx.x;
    if (i < N_NODES) deg[i] = rsqrtf(deg[i]);   // deg buffer becomes dis
}

// ---------------------------------------------------------------------------
// GEMM: H[N,128] = X[N,128] @ W[128,128] using V_WMMA_F32_16X16X4_F32.
// Block = 256 threads = 8 waves; block covers 16 rows x 128 cols.
// X row-band staged in LDS (stride 132 to avoid bank conflicts).
// ISA fp32 fragment layout (wave32):
//   A 16x4 : lanes 0-15 -> M=lane, {K, K+1}; lanes 16-31 -> M=lane-16, {K+2, K+3}
//   B 4x16 : lanes 0-15 -> N=lane, rows {K, K+1}; lanes 16-31 -> rows {K+2, K+3}
//   C/D    : VGPR v, lane l -> M = v + 8*(l>=16), N = l&15
// ---------------------------------------------------------------------------
__global__ __launch_bounds__(256) void gemm128_kernel(const float* __restrict__ X,
                                                      const float* __restrict__ W,
                                                      float* __restrict__ H) {
    __shared__ float lds[16 * 132];
    const int row0 = blockIdx.x * 16;
    const int t = threadIdx.x;

    // Stage 16x128 X tile into LDS (2048 floats, 8 per thread).
#pragma unroll
    for (int i = 0; i < 8; ++i) {
        int idx = t + 256 * i;
        int r = idx >> 7;        // / 128
        int c = idx & 127;
        lds[r * 132 + c] = X[(row0 + r) * DIM + c];
    }
    __syncthreads();

    const int wave = t >> 5;
    const int lane = t & 31;
    const int col0 = wave * 16;
    const int mn = lane & 15;    // M for A frag, N for B/D frags
    const int hi = lane >> 4;    // K-half selector

    v8f acc = {};
#pragma unroll
    for (int k = 0; k < DIM; k += 4) {
        const int k0 = k + 2 * hi;
        v2f a, b;
        a.x = lds[mn * 132 + k0];
        a.y = lds[mn * 132 + k0 + 1];
        b.x = W[k0 * DIM + col0 + mn];
        b.y = W[(k0 + 1) * DIM + col0 + mn];
        // 8 args: (neg_a, A, neg_b, B, c_mod, C, reuse_a, reuse_b)
        acc = __builtin_amdgcn_wmma_f32_16x16x4_f32(
            false, a, false, b, (short)0, acc, false, false);
    }

#pragma unroll
    for (int v = 0; v < 8; ++v) {
        H[(row0 + v + 8 * hi) * DIM + col0 + mn] = acc[v];
    }
}

// ---------------------------------------------------------------------------
// agg[i,j] = dis[i]^2 * h[i,j] + bias[j]   (self-loop term + bias, also the
// initializer the scatter kernel atomically accumulates into)
// ---------------------------------------------------------------------------
__global__ void self_bias_kernel(const float* __restrict__ h,
                                 const float* __restrict__ dis,
                                 const float* __restrict__ bias,
                                 float* __restrict__ agg) {
    int idx = blockIdx.x * 256 + threadIdx.x;   // over N*D, exact multiple
    int i = idx >> 7;
    int j = idx & 127;
    float d = dis[i];
    agg[idx] = d * d * h[idx] + bias[j];
}

// ---------------------------------------------------------------------------
// Edge scatter: one wave per edge; lane handles 4 consecutive features.
// agg[dst] += dis[src]*dis[dst] * h[src]
// ---------------------------------------------------------------------------
__global__ __launch_bounds__(256) void scatter_edges_kernel(const float* __restrict__ h,
                                                            const float* __restrict__ dis,
                                                            const int* __restrict__ src,
                                                            const int* __restrict__ dst,
                                                            float* __restrict__ agg) {
    int e = blockIdx.x * 8 + (threadIdx.x >> 5);
    if (e >= N_EDGES) return;
    int lane = threadIdx.x & 31;
    int s = src[e];
    int d = dst[e];
    float norm = dis[s] * dis[d];
    const float4 v = *(const float4*)(h + s * DIM + lane * 4);
    float* out = agg + d * DIM + lane * 4;
    atomicAdd(out + 0, norm * v.x);
    atomicAdd(out + 1, norm * v.y);
    atomicAdd(out + 2, norm * v.z);
    atomicAdd(out + 3, norm * v.w);
}

// ---------------------------------------------------------------------------
// y = relu(layer_norm(h + ori) * w + b); one wave per row, 4 floats per lane.
// ---------------------------------------------------------------------------
__global__ __launch_bounds__(256) void ln_relu_kernel(const float* __restrict__ h,
                                                      const float* __restrict__ ori,
                                                      const float* __restrict__ w,
                                                      const float* __restrict__ b,
                                                      float* __restrict__ out) {
    int row = blockIdx.x * 8 + (threadIdx.x >> 5);
    int lane = threadIdx.x & 31;
    const float4 hv = *(const float4*)(h + row * DIM + lane * 4);
    const float4 ov = *(const float4*)(ori + row * DIM + lane * 4);
    float4 x;
    x.x = hv.x + ov.x; x.y = hv.y + ov.y; x.z = hv.z + ov.z; x.w = hv.w + ov.w;

    float s  = x.x + x.y + x.z + x.w;
    float sq = x.x * x.x + x.y * x.y + x.z * x.z + x.w * x.w;
#pragma unroll
    for (int m = 16; m >= 1; m >>= 1) {
        s  += __shfl_xor(s, m, 32);
        sq += __shfl_xor(sq, m, 32);
    }
    float mu  = s * (1.0f / 128.0f);
    float var = sq * (1.0f / 128.0f) - mu * mu;
    float r   = rsqrtf(var + LN_EPS);

    const float4 wv = *(const float4*)(w + lane * 4);
    const float4 bv = *(const float4*)(b + lane * 4);
    float4 y;
    y.x = fmaxf((x.x - mu) * r * wv.x + bv.x, 0.0f);
    y.y = fmaxf((x.y - mu) * r * wv.y + bv.y, 0.0f);
    y.z = fmaxf((x.z - mu) * r * wv.z + bv.z, 0.0f);
    y.w = fmaxf((x.w - mu) * r * wv.w + bv.w, 0.0f);
    *(float4*)(out + row * DIM + lane * 4) = y;
}

// ---------------------------------------------------------------------------
extern "C" void kernel_launch(void* const* d_in, const int* in_sizes, int n_in,
                              void* d_out, int out_size, void* d_ws, size_t ws_size,
                              hipStream_t stream) {
    const float* in_feat = (const float*)d_in[0];
    const int*   esrc    = (const int*)d_in[1];
    const int*   edst    = (const int*)d_in[2];
    const float* W1      = (const float*)d_in[3];
    const float* b1      = (const float*)d_in[4];
    const float* W2      = (const float*)d_in[5];
    const float* b2      = (const float*)d_in[6];
    const float* ln1w    = (const float*)d_in[7];
    const float* ln1b    = (const float*)d_in[8];
    const float* ln2w    = (const float*)d_in[9];
    const float* ln2b    = (const float*)d_in[10];
    float* out = (float*)d_out;

    float* ws  = (float*)d_ws;
    float* dis = ws;                                 // N
    float* h   = dis + N_NODES;                      // N*D (GEMM output)
    float* agg = h + (size_t)N_NODES * DIM;          // N*D (conv output)
    float* xb  = agg + (size_t)N_NODES * DIM;        // N*D (LN output)

    const int gN   = (N_NODES + 255) / 256;          // 196
    const int gE   = (N_EDGES + 255) / 256;          // 2344
    const int gND  = (N_NODES * DIM) / 256;          // 25000 exact
    const int gGem = N_NODES / 16;                   // 3125 exact
    const int gSct = N_EDGES / 8;                    // 75000 exact
    const int gLn  = N_NODES / 8;                    // 6250 exact

    // degrees -> dis
    init_deg_kernel<<<gN, 256, 0, stream>>>(dis);
    accum_deg_kernel<<<gE, 256, 0, stream>>>(edst, dis);
    finalize_deg_kernel<<<gN, 256, 0, stream>>>(dis);

    // conv1: h = in_feat @ W1; agg = D^-1/2 (A+I) D^-1/2 h + b1
    gemm128_kernel<<<gGem, 256, 0, stream>>>(in_feat, W1, h);
    self_bias_kernel<<<gND, 256, 0, stream>>>(h, dis, b1, agg);
    scatter_edges_kernel<<<gSct, 256, 0, stream>>>(h, dis, esrc, edst, agg);

    // iter 1: residual + LN1 + relu -> conv2
    ln_relu_kernel<<<gLn, 256, 0, stream>>>(agg, in_feat, ln1w, ln1b, xb);
    gemm128_kernel<<<gGem, 256, 0, stream>>>(xb, W2, h);
    self_bias_kernel<<<gND, 256, 0, stream>>>(h, dis, b2, agg);
    scatter_edges_kernel<<<gSct, 256, 0, stream>>>(h, dis, esrc, edst, agg);

    // iter 2: residual + LN2 + relu -> conv (final output straight into d_out)
    ln_relu_kernel<<<gLn, 256, 0, stream>>>(agg, in_feat, ln2w, ln2b, xb);
    gemm128_kernel<<<gGem, 256, 0, stream>>>(xb, W2, h);
    self_bias_kernel<<<gND, 256, 0, stream>>>(h, dis, b2, out);
    scatter_edges_kernel<<<gSct, 256, 0, stream>>>(h, dis, esrc, edst, out);
}